// InterSO3ConvBlock_4243427689051
// MI455X (gfx1250) — compile-verified
//
#include <hip/hip_runtime.h>
#include <hip/hip_bf16.h>
#include <stddef.h>

// Problem constants (match reference)
#define B_    1
#define C_IN  32
#define C_OUT 64
#define P_    1024
#define NN_   24
#define NA_   60
#define KS_   4
#define CK_   (C_IN * KS_)          // 128, GEMM K
#define PA_   (P_ * NA_)            // 61440, GEMM N
#define SIGMA_ 0.08f
#define EPS_   1e-5f

typedef float v2f __attribute__((ext_vector_type(2)));
typedef float v8f __attribute__((ext_vector_type(8)));

// ---------------------------------------------------------------------------
// Kernel 1: build fk[ck, p*NA + a] = sum_n feats[c, idx[p,n], a] * w[a,k,p,n]
// One block per point p. 256 threads.
// ---------------------------------------------------------------------------
__global__ void __launch_bounds__(256)
fk_build_kernel(const float* __restrict__ xyz,      // [3, P]
                const float* __restrict__ feats,    // [C_IN, P, NA]
                const int*   __restrict__ idx,      // [P, NN]
                const float* __restrict__ anchors,  // [NA, 3, 3]
                const float* __restrict__ kern,     // [KS, 3]
                float*       __restrict__ fk)       // [CK, PA]
{
    __shared__ int   s_idx[NN_];
    __shared__ float s_rel[NN_][3];
    __shared__ float s_r2[NN_];
    __shared__ float s_rk[NA_ * KS_ * 3];   // rk[a,k,i]
    __shared__ float s_rk2[NA_ * KS_];      // |rk|^2
    __shared__ float s_w[NA_ * KS_ * NN_];  // w[a,k,n]

    const int p   = blockIdx.x;
    const int tid = threadIdx.x;

    // --- neighbor relative coordinates -----------------------------------
    if (tid < NN_) {
        const int j = idx[p * NN_ + tid];
        s_idx[tid] = j;
        float r2 = 0.f;
        #pragma unroll
        for (int i = 0; i < 3; ++i) {
            const float r = xyz[i * P_ + j] - xyz[i * P_ + p];
            s_rel[tid][i] = r;
            r2 += r * r;
        }
        s_r2[tid] = r2;
    }

    // --- anchor-rotated kernel points: rk[a,k,i] = sum_j anchors[a,i,j]*kern[k,j]
    for (int e = tid; e < NA_ * KS_; e += 256) {
        const int a = e / KS_;
        const int k = e % KS_;
        float n2 = 0.f;
        #pragma unroll
        for (int i = 0; i < 3; ++i) {
            float v = 0.f;
            #pragma unroll
            for (int j = 0; j < 3; ++j)
                v += anchors[a * 9 + i * 3 + j] * kern[k * 3 + j];
            s_rk[e * 3 + i] = v;
            n2 += v * v;
        }
        s_rk2[e] = n2;
    }
    __syncthreads();

    // --- correlation weights w[a,k,n] = relu(1 - d2/sigma) ----------------
    for (int e = tid; e < NA_ * KS_ * NN_; e += 256) {
        const int a   = e / (KS_ * NN_);
        const int rem = e % (KS_ * NN_);
        const int k   = rem / NN_;
        const int n   = rem % NN_;
        const float* rk = &s_rk[(a * KS_ + k) * 3];
        const float dot = s_rel[n][0] * rk[0] + s_rel[n][1] * rk[1] + s_rel[n][2] * rk[2];
        const float d2  = s_r2[n] + s_rk2[a * KS_ + k] - 2.f * dot;
        const float w   = 1.f - d2 * (1.f / SIGMA_);
        s_w[e] = fmaxf(w, 0.f);
    }
    __syncthreads();

    // --- gather features + aggregate over neighbors -----------------------
    // q = c*NA + a : consecutive threads -> consecutive 'a' -> coalesced feats
    for (int q = tid; q < C_IN * NA_; q += 256) {
        const int c = q / NA_;
        const int a = q % NA_;
        float acc[KS_] = {0.f, 0.f, 0.f, 0.f};
        const float* frow = feats + (size_t)c * (P_ * NA_) + a;
        #pragma unroll 4
        for (int n = 0; n < NN_; ++n) {
            const float f = frow[(size_t)s_idx[n] * NA_];
            const float* wn = &s_w[a * (KS_ * NN_) + n];
            #pragma unroll
            for (int k = 0; k < KS_; ++k)
                acc[k] += f * wn[k * NN_];
        }
        const size_t pa = (size_t)p * NA_ + a;
        #pragma unroll
        for (int k = 0; k < KS_; ++k)
            fk[(size_t)(c * KS_ + k) * PA_ + pa] = acc[k];
    }
}

// ---------------------------------------------------------------------------
// Kernel 2: out[d, pa] = sum_ck W[d,ck] * fk[ck,pa]  via V_WMMA_F32_16X16X4_F32
// Each wave32 owns one 16-wide N-tile and all 4 M-tiles (C_OUT=64).
// A 16x4 f32 layout : VGPR0 = {K=0 | K=2}, VGPR1 = {K=1 | K=3} across lane halves
// B 4x16 f32 layout : mirrored (N across lanes 0-15 / 16-31)
// D 16x16 f32 layout: VGPR r = {M=r | M=r+8} across lane halves
// ---------------------------------------------------------------------------
__global__ void __launch_bounds__(256)
gemm_wmma_kernel(const float* __restrict__ Wm,   // [C_OUT, CK]
                 const float* __restrict__ fk,   // [CK, PA]
                 float*       __restrict__ out)  // [C_OUT, PA]
{
    const int tid   = threadIdx.x;
    const int wave  = tid >> 5;
    const int lane  = tid & 31;
    const int lh    = lane & 15;
    const int hi    = lane >> 4;          // 0: lanes 0-15, 1: lanes 16-31
    const int ntile = blockIdx.x * 8 + wave;
    const int n     = ntile * 16 + lh;    // global pa column

    v8f acc0 = {}, acc1 = {}, acc2 = {}, acc3 = {};

    for (int k0 = 0; k0 < CK_; k0 += 4) {
        const int kk = k0 + hi * 2;

        // B operand: fk[kk..kk+1, n]
        v2f b;
        b.x = fk[(size_t)kk * PA_ + n];
        b.y = fk[(size_t)(kk + 1) * PA_ + n];

        if (k0 + 4 < CK_)
            __builtin_prefetch(&fk[(size_t)(kk + 4) * PA_ + n], 0, 3);

        // A operands for the four 16-row strips of W
        v2f a0, a1, a2, a3;
        a0.x = Wm[(size_t)( 0 + lh) * CK_ + kk]; a0.y = Wm[(size_t)( 0 + lh) * CK_ + kk + 1];
        a1.x = Wm[(size_t)(16 + lh) * CK_ + kk]; a1.y = Wm[(size_t)(16 + lh) * CK_ + kk + 1];
        a2.x = Wm[(size_t)(32 + lh) * CK_ + kk]; a2.y = Wm[(size_t)(32 + lh) * CK_ + kk + 1];
        a3.x = Wm[(size_t)(48 + lh) * CK_ + kk]; a3.y = Wm[(size_t)(48 + lh) * CK_ + kk + 1];

        acc0 = __builtin_amdgcn_wmma_f32_16x16x4_f32(false, a0, false, b, (short)0, acc0, false, false);
        acc1 = __builtin_amdgcn_wmma_f32_16x16x4_f32(false, a1, false, b, (short)0, acc1, false, false);
        acc2 = __builtin_amdgcn_wmma_f32_16x16x4_f32(false, a2, false, b, (short)0, acc2, false, false);
        acc3 = __builtin_amdgcn_wmma_f32_16x16x4_f32(false, a3, false, b, (short)0, acc3, false, false);
    }

    // store D tiles: VGPR r holds (M = r + hi*8, N = n)
    #pragma unroll
    for (int r = 0; r < 8; ++r) {
        const int m = r + hi * 8;
        out[(size_t)( 0 + m) * PA_ + n] = acc0[r];
        out[(size_t)(16 + m) * PA_ + n] = acc1[r];
        out[(size_t)(32 + m) * PA_ + n] = acc2[r];
        out[(size_t)(48 + m) * PA_ + n] = acc3[r];
    }
}

// ---------------------------------------------------------------------------
// Kernel 3: per-channel mean / inv-std (deterministic tree reduction)
// One block per channel.
// ---------------------------------------------------------------------------
__global__ void __launch_bounds__(256)
stats_kernel(const float* __restrict__ out, float* __restrict__ mu, float* __restrict__ inv)
{
    __shared__ float ssum[256];
    __shared__ float ssq[256];
    const int ch  = blockIdx.x;
    const int tid = threadIdx.x;
    float s = 0.f, q = 0.f;
    const float* row = out + (size_t)ch * PA_;
    for (int i = tid; i < PA_; i += 256) {
        const float v = row[i];
        s += v;
        q += v * v;
    }
    ssum[tid] = s;
    ssq[tid]  = q;
    __syncthreads();
    for (int off = 128; off > 0; off >>= 1) {
        if (tid < off) {
            ssum[tid] += ssum[tid + off];
            ssq[tid]  += ssq[tid + off];
        }
        __syncthreads();
    }
    if (tid == 0) {
        const float m = ssum[0] * (1.f / (float)PA_);
        const float v = ssq[0] * (1.f / (float)PA_) - m * m;
        mu[ch]  = m;
        inv[ch] = rsqrtf(v + EPS_);
    }
}

// ---------------------------------------------------------------------------
// Kernel 4: in-place instance-norm + ReLU
// ---------------------------------------------------------------------------
__global__ void __launch_bounds__(256)
norm_relu_kernel(float* __restrict__ out, const float* __restrict__ mu,
                 const float* __restrict__ inv)
{
    const size_t i = (size_t)blockIdx.x * blockDim.x + threadIdx.x;
    if (i < (size_t)C_OUT * PA_) {
        const int ch = (int)(i / PA_);
        const float y = (out[i] - mu[ch]) * inv[ch];
        out[i] = fmaxf(y, 0.f);
    }
}

// ---------------------------------------------------------------------------
extern "C" void kernel_launch(void* const* d_in, const int* in_sizes, int n_in,
                              void* d_out, int out_size, void* d_ws, size_t ws_size,
                              hipStream_t stream) {
    const float* xyz     = (const float*)d_in[0];  // [1,3,1024]
    const float* feats   = (const float*)d_in[1];  // [1,32,1024,60]
    const int*   idx     = (const int*)  d_in[2];  // [1,1024,24]
    const float* anchors = (const float*)d_in[3];  // [60,3,3]
    const float* kern    = (const float*)d_in[4];  // [4,3]
    const float* Wm      = (const float*)d_in[5];  // [64,128]
    float* out = (float*)d_out;                    // [1,64,1024,60]

    // workspace: fk [128, 61440] floats (31.46 MB) + 128 floats of stats
    float* fk = (float*)d_ws;
    float* mu = fk + (size_t)CK_ * PA_;
    float* inv = mu + C_OUT;

    fk_build_kernel<<<P_, 256, 0, stream>>>(xyz, feats, idx, anchors, kern, fk);

    // N-tiles = 61440/16 = 3840; 8 waves/block -> 480 blocks
    gemm_wmma_kernel<<<(PA_ / 16) / 8, 256, 0, stream>>>(Wm, fk, out);

    stats_kernel<<<C_OUT, 256, 0, stream>>>(out, mu, inv);

    const size_t total = (size_t)C_OUT * PA_;
    norm_relu_kernel<<<(unsigned)((total + 255) / 256), 256, 0, stream>>>(out, mu, inv);
}